// Tacotron2Loss_30374008717519
// MI455X (gfx1250) — compile-verified
//
#include <hip/hip_runtime.h>
#include <hip/hip_bf16.h>

// Problem constants (match reference)
#define B_    64
#define T_    2000
#define L_    400
#define M_    80
#define EPS_  1e-8f
#define BT_   (B_ * T_)
#define NMEL_ (B_ * T_ * M_)
#define NMEL4_ (NMEL_ / 4)     // 2,560,000 float4s; 80 % 4 == 0 -> quads never cross (b,t) rows
#define NQ_ROW_ (M_ / 4)       // 20 quads per (b,t) mel row
#define BT4_  (BT_ / 4)        // gate quads
#define LQ_   (L_ / 4)         // 100 quads per alignment row

// Launch geometry
#define NTHREADS 256
#define NB_MEL   1024
#define NB_GATE  64
#define NB_ATTN  2000
#define WAVES_ATTN ((NB_ATTN * NTHREADS) / 32)

// Workspace float offsets
#define OFF_MEL1 0
#define OFF_MEL2 (OFF_MEL1 + NB_MEL)
#define OFF_GATE (OFF_MEL2 + NB_MEL)
#define OFF_KL   (OFF_GATE + NB_GATE)
#define OFF_ENT  (OFF_KL + NB_ATTN)

typedef __attribute__((ext_vector_type(2))) float v2f;
typedef __attribute__((ext_vector_type(4))) float v4f;
typedef __attribute__((ext_vector_type(8))) float v8f;

// Exact f32 sum of the 32 lane values; result broadcast to all lanes.
// Stage 1: A holds data (A[m,0]=v_m, A[m,2]=v_{m+16}, .y=0), B = ones
//   => D[m,n] = v_m + v_{m+16}. Per-lane sum of the 8 D VGPRs gives
//   S_lo (lanes 0-15) / S_hi (lanes 16-31).
// Stage 2: A = ones, B.x = that value => every column sums one lo-half and
//   one hi-half contribution => D2 = S_lo + S_hi = total, in every lane.
__device__ __forceinline__ float wave_reduce_sum(float v) {
    v2f a;    a[0] = v;    a[1] = 0.0f;
    v2f ones; ones[0] = 1.0f; ones[1] = 1.0f;
    v8f c = {};
    v8f d1 = __builtin_amdgcn_wmma_f32_16x16x4_f32(
        false, a, false, ones, (short)0, c, false, false);
    float t = ((d1[0] + d1[1]) + (d1[2] + d1[3])) +
              ((d1[4] + d1[5]) + (d1[6] + d1[7]));
    v2f b2; b2[0] = t; b2[1] = 0.0f;
    v8f d2 = __builtin_amdgcn_wmma_f32_16x16x4_f32(
        false, ones, false, b2, (short)0, c, false, false);
    return d2[0];
}

// Block (256 threads = 8 waves) exact sum; result returned to all threads.
__device__ __forceinline__ float block_reduce_256(float v, float* s8) {
    float w = wave_reduce_sum(v);
    int wid = threadIdx.x >> 5;
    if ((threadIdx.x & 31) == 0) s8[wid] = w;
    __syncthreads();
    float r = ((s8[0] + s8[1]) + (s8[2] + s8[3])) +
              ((s8[4] + s8[5]) + (s8[6] + s8[7]));
    __syncthreads();
    return r;
}

// ---------------- mel L1 partials (float4 / global_load_b128) ----------------
__global__ __launch_bounds__(NTHREADS)
void mel_partial_kernel(const float* __restrict__ postnet,
                        const float* __restrict__ mel_out,
                        const float* __restrict__ mel_tgt,
                        const int*   __restrict__ mel_len,
                        float* __restrict__ p1, float* __restrict__ p2) {
    __shared__ float s8[8];
    const v4f* __restrict__ po4 = (const v4f*)postnet;
    const v4f* __restrict__ mo4 = (const v4f*)mel_out;
    const v4f* __restrict__ mt4 = (const v4f*)mel_tgt;
    float a1 = 0.0f, a2 = 0.0f;
    int stride = gridDim.x * blockDim.x;
    for (int q = blockIdx.x * blockDim.x + threadIdx.x; q < NMEL4_; q += stride) {
        int t = (q / NQ_ROW_) % T_;
        int b = q / (NQ_ROW_ * T_);
        float vf = (t < mel_len[b]) ? 1.0f : 0.0f;
        v4f o = mo4[q];
        v4f p = po4[q];
        v4f g = mt4[q];
        float d1 = fabsf(o[0] - g[0]) + fabsf(o[1] - g[1]) +
                   fabsf(o[2] - g[2]) + fabsf(o[3] - g[3]);
        float d2 = fabsf(p[0] - g[0]) + fabsf(p[1] - g[1]) +
                   fabsf(p[2] - g[2]) + fabsf(p[3] - g[3]);
        a1 = fmaf(d1, vf, a1);
        a2 = fmaf(d2, vf, a2);
    }
    float s1 = block_reduce_256(a1, s8);
    float s2 = block_reduce_256(a2, s8);
    if (threadIdx.x == 0) { p1[blockIdx.x] = s1; p2[blockIdx.x] = s2; }
}

// ---------------- gate BCE partials (float4) ----------------
__global__ __launch_bounds__(NTHREADS)
void gate_partial_kernel(const float* __restrict__ gate_out,
                         const float* __restrict__ gate_tgt,
                         float* __restrict__ pg) {
    __shared__ float s8[8];
    const v4f* __restrict__ go4 = (const v4f*)gate_out;
    const v4f* __restrict__ gt4 = (const v4f*)gate_tgt;
    float acc = 0.0f;
    int stride = gridDim.x * blockDim.x;
    for (int q = blockIdx.x * blockDim.x + threadIdx.x; q < BT4_; q += stride) {
        v4f x4 = go4[q];
        v4f z4 = gt4[q];
        #pragma unroll
        for (int j = 0; j < 4; ++j) {
            float x = x4[j], z = z4[j];
            acc += fmaxf(x, 0.0f) - x * z + log1pf(expf(-fabsf(x)));
        }
    }
    float s = block_reduce_256(acc, s8);
    if (threadIdx.x == 0) pg[blockIdx.x] = s;
}

// ---------------- attention KL + entropy partials ----------------
// One wave per (b,t) row; per-row Gaussian target rowsum is reduced and
// broadcast with the WMMA wave reducer. Alignments streamed as float4.
__global__ __launch_bounds__(NTHREADS)
void attn_partial_kernel(const float* __restrict__ align,
                         const int*   __restrict__ text_len,
                         float* __restrict__ pkl, float* __restrict__ pent) {
    __shared__ float s8[8];
    int lane  = threadIdx.x & 31;
    int gwave = (blockIdx.x * blockDim.x + threadIdx.x) >> 5;
    float kl_acc = 0.0f, ent_acc = 0.0f;

    for (int row = gwave; row < BT_; row += WAVES_ATTN) {
        int b = row / T_;
        int t = row - b * T_;
        int Lb = text_len[b];
        float sigma = fminf(fmaxf((float)Lb * 0.05f, 3.0f), 10.0f);
        float inv_sigma = 1.0f / sigma;
        int ep_i = (t * Lb) / T_;
        if (ep_i > Lb - 1) ep_i = Lb - 1;
        float exp_pos = (float)ep_i;
        long base = (long)row * L_;
        const v4f* __restrict__ arow = (const v4f*)(align + base);

        // pass 1: Gaussian row sum (no memory traffic)
        float gsum = 0.0f;
        for (int l = lane; l < L_; l += 32) {
            float z = ((float)l - exp_pos) * inv_sigma;
            float g = (l < Lb) ? expf(-0.5f * z * z) : 0.0f;
            gsum += g;
        }
        float rowsum = wave_reduce_sum(gsum);   // broadcast to all lanes
        float inv = 1.0f / (rowsum + EPS_);

        // pass 2: KL + entropy; one float4 per lane-iteration (b128 loads)
        for (int q = lane; q < LQ_; q += 32) {
            __builtin_prefetch((const float*)(arow + q) + 1024, 0, 0);
            v4f a4 = arow[q];
            #pragma unroll
            for (int j = 0; j < 4; ++j) {
                int l = 4 * q + j;
                float as = fmaxf(a4[j], EPS_);
                float lp = logf(as);
                ent_acc  = fmaf(-as, lp, ent_acc);
                float z  = ((float)l - exp_pos) * inv_sigma;
                float g  = (l < Lb) ? expf(-0.5f * z * z) : 0.0f;
                float tgt = g * inv;
                float lt  = logf(fmaxf(tgt, EPS_));
                kl_acc += (g > 0.0f) ? tgt * (lt - lp) : 0.0f;
            }
        }
    }
    float skl = block_reduce_256(kl_acc, s8);
    float sent = block_reduce_256(ent_acc, s8);
    if (threadIdx.x == 0) { pkl[blockIdx.x] = skl; pent[blockIdx.x] = sent; }
}

// ---------------- finalize ----------------
__global__ __launch_bounds__(NTHREADS)
void finalize_kernel(const float* __restrict__ ws,
                     const int* __restrict__ mel_len,
                     float* __restrict__ out) {
    __shared__ float s8[8];
    int tid = threadIdx.x;

    float p = 0.0f;
    for (int i = tid; i < NB_MEL; i += NTHREADS) p += ws[OFF_MEL1 + i];
    float mel1 = block_reduce_256(p, s8);

    p = 0.0f;
    for (int i = tid; i < NB_MEL; i += NTHREADS) p += ws[OFF_MEL2 + i];
    float mel2 = block_reduce_256(p, s8);

    p = 0.0f;
    for (int i = tid; i < NB_GATE; i += NTHREADS) p += ws[OFF_GATE + i];
    float gsum = block_reduce_256(p, s8);

    p = 0.0f;
    for (int i = tid; i < NB_ATTN; i += NTHREADS) p += ws[OFF_KL + i];
    float klsum = block_reduce_256(p, s8);

    p = 0.0f;
    for (int i = tid; i < NB_ATTN; i += NTHREADS) p += ws[OFF_ENT + i];
    float entsum = block_reduce_256(p, s8);

    p = 0.0f;
    for (int i = tid; i < B_; i += NTHREADS) {
        int v = mel_len[i];
        p += (float)(v < T_ ? v : T_);
    }
    float vcount = block_reduce_256(p, s8);

    if (tid == 0) {
        float n_valid   = vcount * (float)M_;
        float loss_mel  = mel1 / n_valid + mel2 / n_valid;
        float loss_gate = gsum / (float)BT_;
        float kl        = fminf(klsum / (float)BT_, 150.0f);
        float ent       = entsum / (float)BT_;
        float ratio     = fmaxf(ent / 3.5f, 0.0f);
        float weight    = (ent <= 3.5f) ? fmaxf(0.2f, ratio) : 1.0f;
        out[0] = loss_mel + loss_gate + weight * kl;
        out[1] = loss_mel;
        out[2] = loss_gate;
        out[3] = kl;
    }
}

extern "C" void kernel_launch(void* const* d_in, const int* in_sizes, int n_in,
                              void* d_out, int out_size, void* d_ws, size_t ws_size,
                              hipStream_t stream) {
    const float* postnet  = (const float*)d_in[0];
    const float* mel_out  = (const float*)d_in[1];
    const float* gate_out = (const float*)d_in[2];
    const float* align    = (const float*)d_in[3];
    const float* mel_tgt  = (const float*)d_in[4];
    const float* gate_tgt = (const float*)d_in[5];
    const int*   mel_len  = (const int*)d_in[6];
    const int*   text_len = (const int*)d_in[7];
    float* ws  = (float*)d_ws;
    float* out = (float*)d_out;

    mel_partial_kernel<<<NB_MEL, NTHREADS, 0, stream>>>(
        postnet, mel_out, mel_tgt, mel_len, ws + OFF_MEL1, ws + OFF_MEL2);
    gate_partial_kernel<<<NB_GATE, NTHREADS, 0, stream>>>(
        gate_out, gate_tgt, ws + OFF_GATE);
    attn_partial_kernel<<<NB_ATTN, NTHREADS, 0, stream>>>(
        align, text_len, ws + OFF_KL, ws + OFF_ENT);
    finalize_kernel<<<1, NTHREADS, 0, stream>>>(ws, mel_len, out);
}